// Myloss_15461882265979
// MI455X (gfx1250) — compile-verified
//
#include <hip/hip_runtime.h>
#include <cstdint>

// ---------------------------------------------------------------------------
// Focal-style loss over [N=262144, C=1000] logits.
// Memory-bound: 1.05 GB single-use stream -> ~45us floor at 23.3 TB/s.
// Strategy: per-wave double-buffered gfx1250 async global->LDS row staging
// (ASYNCcnt / s_wait_asynccnt), wave32 butterfly reductions for max/top2/sum.
// ---------------------------------------------------------------------------

#define C_CLASSES 1000
#define C4 250                 // float4 chunks per row (1000 floats, 16B-aligned rows)
#define WAVES_PER_BLOCK 8
#define BLOCK_THREADS 256
#define NBLOCKS 1024

// The gfx1250 async-load builtin is typed over int4 pointers:
//   void __builtin_amdgcn_global_load_async_to_lds_b128(
//       global v4i* gaddr, lds v4i* ldsaddr, imm int offset, imm int cpol)
typedef int v4i __attribute__((vector_size(16)));
typedef __attribute__((address_space(1))) v4i* g4p_t;
typedef __attribute__((address_space(3))) v4i* l4p_t;

// One 16-byte async DMA chunk: global -> LDS, tracked by ASYNCcnt.
__device__ __forceinline__ void async_copy16(const float4* g, const float4* l) {
#if __has_builtin(__builtin_amdgcn_global_load_async_to_lds_b128)
  g4p_t gp = (g4p_t)(unsigned long long)(uintptr_t)g;
  l4p_t lp = (l4p_t)(unsigned)(uintptr_t)l;   // low 32 bits of flat = LDS offset
  __builtin_amdgcn_global_load_async_to_lds_b128(gp, lp, 0, 0);
#else
  unsigned           loff = (unsigned)(uintptr_t)l;
  unsigned long long ga   = (unsigned long long)(uintptr_t)g;
  asm volatile("global_load_async_to_lds_b128 %0, %1, off"
               :: "v"(loff), "v"(ga) : "memory");
#endif
}

#if __has_builtin(__builtin_amdgcn_s_wait_asynccnt)
#define WAIT_ASYNC(N_) do { __builtin_amdgcn_s_wait_asynccnt(N_); \
                            asm volatile("" ::: "memory"); } while (0)
#else
#define WAIT_ASYNC(N_) asm volatile("s_wait_asynccnt %0" :: "i"(N_) : "memory")
#endif

// Issue exactly 8 async B128 ops covering one row (index clamped so the
// ASYNCcnt bookkeeping is EXEC-independent: lanes 26..31 at k=7 redundantly
// re-copy chunk 249 — same data, same destination, harmless).
__device__ __forceinline__ void issue_row_copy(const float4* src, float4* dst, int lane) {
#pragma unroll
  for (int k = 0; k < 8; ++k) {
    int i = lane + (k << 5);
    if (i > C4 - 1) i = C4 - 1;
    async_copy16(src + i, dst + i);
  }
}

__global__ __launch_bounds__(BLOCK_THREADS)
void focal_rows_kernel(const float* __restrict__ pred,
                       const int*   __restrict__ labels,
                       float*       __restrict__ partial,
                       int N) {
  __shared__ float4 buf[2][WAVES_PER_BLOCK][C4];   // 64000 B
  __shared__ float  swacc[WAVES_PER_BLOCK];

  const int lane  = threadIdx.x & 31;
  const int wave  = threadIdx.x >> 5;
  const int gwave = blockIdx.x * WAVES_PER_BLOCK + wave;
  const int totw  = gridDim.x * WAVES_PER_BLOCK;
  const float4* gp = (const float4*)pred;
  const float NINF = -__builtin_inff();

  float acc = 0.0f;
  int   cur = 0;
  int   row = gwave;

  if (row < N) issue_row_copy(gp + (size_t)row * C4, &buf[0][wave][0], lane);

  for (; row < N; row += totw) {
    const int nxt = row + totw;
    if (nxt < N) {
      issue_row_copy(gp + (size_t)nxt * C4, &buf[cur ^ 1][wave][0], lane);
      WAIT_ASYNC(8);   // <=8 outstanding: current row landed, prefetch in flight
    } else {
      WAIT_ASYNC(0);
    }

    const float4* lb = &buf[cur][wave][0];

    // One LDS sweep into registers (8 x ds_load_b128 per lane).
    float4 v[8];
#pragma unroll
    for (int k = 0; k < 8; ++k) {
      const int i = lane + (k << 5);
      if (i < C4) {
        v[k] = lb[i];
      } else {
        v[k].x = NINF; v[k].y = NINF; v[k].z = NINF; v[k].w = NINF;
      }
    }

    // Lane-local top-2 (branchless; ties behave like top_k).
    float m1 = NINF, m2 = NINF;
#pragma unroll
    for (int k = 0; k < 8; ++k) {
      const float e[4] = {v[k].x, v[k].y, v[k].z, v[k].w};
#pragma unroll
      for (int j = 0; j < 4; ++j) {
        const float hi = fmaxf(m1, e[j]);
        const float lo = fminf(m1, e[j]);
        m1 = hi;
        m2 = fmaxf(m2, lo);
      }
    }
    // Wave32 butterfly merge of (m1, m2) pairs.
#pragma unroll
    for (int off = 16; off > 0; off >>= 1) {
      const float o1 = __shfl_xor(m1, off, 32);
      const float o2 = __shfl_xor(m2, off, 32);
      const float hi = fmaxf(m1, o1);
      const float lo = fminf(m1, o1);
      m1 = hi;
      m2 = fmaxf(fmaxf(m2, o2), lo);
    }

    // Sum of exp(x - m1); masked slots are -inf -> exp == 0.
    float s = 0.0f;
#pragma unroll
    for (int k = 0; k < 8; ++k) {
      s += __expf(v[k].x - m1);
      s += __expf(v[k].y - m1);
      s += __expf(v[k].z - m1);
      s += __expf(v[k].w - m1);
    }
#pragma unroll
    for (int off = 16; off > 0; off >>= 1) s += __shfl_xor(s, off, 32);

    // Per-row loss (all lanes compute the same value; lane 0 accumulates).
    const int   lbl = labels[row];
    const float xl  = ((const float*)lb)[lbl];
    const float p   = __expf(xl - m1) / s;
    const float diff = (1.0f - __expf(m2 - m1)) / s;   // (p_top1 - p_top2)
    float y = 3.0f * diff;
    if (y <= 0.01f) y = 0.0f;
    const float focal = (y == 0.0f) ? 1.0f : __powf(1.0f - p, y);
    const float li = -focal * __logf(p);
    if (lane == 0) acc += li;

    cur ^= 1;
  }

  if (lane == 0) swacc[wave] = acc;
  __syncthreads();
  if (threadIdx.x == 0) {
    float b = 0.0f;
#pragma unroll
    for (int w = 0; w < WAVES_PER_BLOCK; ++w) b += swacc[w];
    partial[blockIdx.x] = b;
  }
}

__global__ __launch_bounds__(256)
void final_reduce_kernel(const float* __restrict__ partial, int n,
                         float* __restrict__ out, float invN) {
  __shared__ float sh[256];
  float s = 0.0f;
  for (int i = threadIdx.x; i < n; i += 256) s += partial[i];
  sh[threadIdx.x] = s;
  __syncthreads();
  for (int w = 128; w > 0; w >>= 1) {
    if ((int)threadIdx.x < w) sh[threadIdx.x] += sh[threadIdx.x + w];
    __syncthreads();
  }
  if (threadIdx.x == 0) out[0] = sh[0] * invN;
}

extern "C" void kernel_launch(void* const* d_in, const int* in_sizes, int n_in,
                              void* d_out, int out_size, void* d_ws, size_t ws_size,
                              hipStream_t stream) {
  (void)n_in; (void)out_size; (void)ws_size;
  const float* pred   = (const float*)d_in[0];
  const int*   labels = (const int*)d_in[1];     // JAX default (no x64) => int32
  const int    N      = in_sizes[1];
  float*       partial = (float*)d_ws;           // NBLOCKS floats of scratch

  focal_rows_kernel<<<NBLOCKS, BLOCK_THREADS, 0, stream>>>(pred, labels, partial, N);
  final_reduce_kernel<<<1, 256, 0, stream>>>(partial, NBLOCKS, (float*)d_out,
                                             1.0f / (float)N);
}